// GaussianWindow_73272142069912
// MI455X (gfx1250) — compile-verified
//
#include <hip/hip_runtime.h>

// Problem dims (fixed by reference)
#define T_STEPS 4096
#define D_HID   1024
#define K_COMP  32
#define N3K     96          // 3*K
#define U_WIN   1023        // window_size - 1
#define U_PAD   1024        // padded K-dim for WMMA GEMM2
#define V_VOCAB 128

typedef __attribute__((ext_vector_type(2))) float v2f;
typedef __attribute__((ext_vector_type(8))) float v8f;

// ---------------------------------------------------------------------------
// Stage 1: abk = exp(x @ W + b)    [T, 96]   via V_WMMA_F32_16X16X4_F32
// One wave (32 lanes) owns one 16x16 output tile; 4 waves per block.
// ---------------------------------------------------------------------------
__global__ __launch_bounds__(128) void gw_gemm_abk(const float* __restrict__ x,
                                                   const float* __restrict__ W,
                                                   const float* __restrict__ bias,
                                                   float* __restrict__ abk) {
  const int lane = threadIdx.x & 31;
  const int wave = threadIdx.x >> 5;
  const int tile = blockIdx.x * 4 + wave;
  const int ntn  = N3K / 16;             // 6 tiles along N
  const int tm   = tile / ntn;           // 0..255
  const int tn   = tile % ntn;           // 0..5
  if (tm >= T_STEPS / 16) return;

  const int half = lane >> 4;            // 0: K={0,1}  1: K={2,3}
  const int l15  = lane & 15;
  const int arow = tm * 16 + l15;        // A-matrix row (M) carried by this lane
  const int bcol = tn * 16 + l15;        // B/C/D column (N) carried by this lane

  v8f c = {};
  #pragma unroll 4
  for (int k0 = 0; k0 < D_HID; k0 += 4) {
    const int ka = k0 + half * 2;
    v2f a, b;
    a.x = x[arow * D_HID + ka];
    a.y = x[arow * D_HID + ka + 1];
    b.x = W[(ka    ) * N3K + bcol];
    b.y = W[(ka + 1) * N3K + bcol];
    c = __builtin_amdgcn_wmma_f32_16x16x4_f32(false, a, false, b,
                                              (short)0, c, false, false);
  }

  const float bv = bias[bcol];
  #pragma unroll
  for (int i = 0; i < 8; ++i) {
    const int r = tm * 16 + i + half * 8;   // C/D: vgpr i -> row i (+8 for hi half)
    abk[r * N3K + bcol] = __expf(c[i] + bv);
  }
}

// ---------------------------------------------------------------------------
// Stage 2: kappa_cs[t,k] = cumsum_t(abk[t, 64+k]); one block per component k.
// Thread-local serial prefix (16 elems) + Hillis-Steele block scan in LDS.
// ---------------------------------------------------------------------------
__global__ __launch_bounds__(256) void gw_cumsum(const float* __restrict__ abk,
                                                 float* __restrict__ kappa_cs) {
  const int k   = blockIdx.x;            // 0..31
  const int tid = threadIdx.x;           // 0..255
  const int PER = T_STEPS / 256;         // 16 rows per thread

  float vals[16];
  float run = 0.f;
  const int base = tid * PER;
  #pragma unroll
  for (int i = 0; i < PER; ++i) {
    run += abk[(base + i) * N3K + 64 + k];
    vals[i] = run;                       // inclusive within-thread prefix
  }

  __shared__ float s[256];
  s[tid] = run;
  __syncthreads();
  #pragma unroll
  for (int off = 1; off < 256; off <<= 1) {
    const float v = (tid >= off) ? s[tid - off] : 0.f;
    __syncthreads();
    s[tid] += v;
    __syncthreads();
  }
  const float prefix = s[tid] - run;     // exclusive prefix of thread totals

  #pragma unroll
  for (int i = 0; i < PER; ++i)
    kappa_cs[(base + i) * K_COMP + k] = prefix + vals[i];
}

// ---------------------------------------------------------------------------
// Stage 3: phi[t,u] = sum_k alpha[t,k] * exp(-beta[t,k] * (kappa[t,k]-u)^2)
// One block per timestep t; mixture params staged in LDS; 4 u's per lane.
// phi padded to U_PAD columns (col 1023 = 0).
// ---------------------------------------------------------------------------
__global__ __launch_bounds__(256) void gw_phi(const float* __restrict__ abk,
                                              const float* __restrict__ kappa_cs,
                                              float* __restrict__ phi) {
  const int t = blockIdx.x;
  __shared__ float sa[K_COMP], sb[K_COMP], sk[K_COMP];
  if (threadIdx.x < K_COMP) {
    sa[threadIdx.x] = abk[t * N3K + threadIdx.x];            // alpha
    sb[threadIdx.x] = abk[t * N3K + 32 + threadIdx.x];       // beta
    sk[threadIdx.x] = kappa_cs[t * K_COMP + threadIdx.x];    // kappa (cumsum'd)
  }
  __syncthreads();

  #pragma unroll
  for (int j = 0; j < U_PAD / 256; ++j) {
    const int u = threadIdx.x + 256 * j;
    const float uf = (float)u;
    float acc = 0.f;
    #pragma unroll 8
    for (int k = 0; k < K_COMP; ++k) {
      const float d = sk[k] - uf;
      acc += sa[k] * __expf(-sb[k] * d * d);
    }
    phi[t * U_PAD + u] = (u < U_WIN) ? acc : 0.f;
  }
}

// ---------------------------------------------------------------------------
// Stage 4: window = phi[T, U_PAD] @ onehot[U_WIN, V]  (K row 1023 -> zeros)
// via V_WMMA_F32_16X16X4_F32, one wave per 16x16 tile.
// ---------------------------------------------------------------------------
__global__ __launch_bounds__(128) void gw_gemm_win(const float* __restrict__ phi,
                                                   const float* __restrict__ onehot,
                                                   float* __restrict__ out) {
  const int lane = threadIdx.x & 31;
  const int wave = threadIdx.x >> 5;
  const int tile = blockIdx.x * 4 + wave;
  const int ntn  = V_VOCAB / 16;         // 8 tiles along N
  const int tm   = tile / ntn;
  const int tn   = tile % ntn;
  if (tm >= T_STEPS / 16) return;

  const int half = lane >> 4;
  const int l15  = lane & 15;
  const int arow = tm * 16 + l15;
  const int bcol = tn * 16 + l15;

  v8f c = {};
  #pragma unroll 4
  for (int k0 = 0; k0 < U_PAD; k0 += 4) {
    const int ka = k0 + half * 2;
    v2f a, b;
    a.x = phi[arow * U_PAD + ka];
    a.y = phi[arow * U_PAD + ka + 1];
    b.x = (ka     < U_WIN) ? onehot[(ka    ) * V_VOCAB + bcol] : 0.f;
    b.y = (ka + 1 < U_WIN) ? onehot[(ka + 1) * V_VOCAB + bcol] : 0.f;
    c = __builtin_amdgcn_wmma_f32_16x16x4_f32(false, a, false, b,
                                              (short)0, c, false, false);
  }

  #pragma unroll
  for (int i = 0; i < 8; ++i) {
    const int r = tm * 16 + i + half * 8;
    out[r * V_VOCAB + bcol] = c[i];
  }
}

// ---------------------------------------------------------------------------
extern "C" void kernel_launch(void* const* d_in, const int* in_sizes, int n_in,
                              void* d_out, int out_size, void* d_ws, size_t ws_size,
                              hipStream_t stream) {
  // setup_inputs() order: x [T,D], onehot [U_WIN,V], W [D,3K], b [3K]
  const float* x      = (const float*)d_in[0];
  const float* onehot = (const float*)d_in[1];
  const float* W      = (const float*)d_in[2];
  const float* b      = (const float*)d_in[3];
  float* out = (float*)d_out;

  // Workspace carve-up (all f32):
  //   abk      : T * 96        =  1.57 MB
  //   kappa_cs : T * 32        =  0.52 MB
  //   phi      : T * U_PAD     = 16.78 MB
  float* abk      = (float*)d_ws;
  float* kappa_cs = abk + (size_t)T_STEPS * N3K;
  float* phi      = kappa_cs + (size_t)T_STEPS * K_COMP;

  // Stage 1: 256 * 6 = 1536 tiles, 4 waves/block -> 384 blocks
  gw_gemm_abk<<<(T_STEPS / 16) * (N3K / 16) / 4, 128, 0, stream>>>(x, W, b, abk);

  // Stage 2: one block per mixture component
  gw_cumsum<<<K_COMP, 256, 0, stream>>>(abk, kappa_cs);

  // Stage 3: one block per timestep
  gw_phi<<<T_STEPS, 256, 0, stream>>>(abk, kappa_cs, phi);

  // Stage 4: 256 * 8 = 2048 tiles, 4 waves/block -> 512 blocks
  gw_gemm_win<<<(T_STEPS / 16) * (V_VOCAB / 16) / 4, 128, 0, stream>>>(phi, onehot, out);
}